// RoIAlign_64879775973770
// MI455X (gfx1250) — compile-verified
//
#include <hip/hip_runtime.h>

// Problem constants (match reference).
#define CHN 256
#define FH  256
#define FW  256
#define RS  7
#define BINS_PER_ROI (RS * RS)   // 49

// Wave32 max-reduction (gfx1250 is wave32-only; lowers to ds_swizzle/permlane).
__device__ __forceinline__ float wave32_max(float v) {
#pragma unroll
    for (int off = 16; off > 0; off >>= 1)
        v = fmaxf(v, __shfl_xor(v, off, 32));
    return v;
}

// Precompute per-sample corner offsets & weights exactly like the reference:
// clamp indices FIRST, then weights from clamped indices (so clamped-equal
// rows/cols yield weight sum 0 -> value 0, matching JAX).
struct Sample { int o11, o12, o21, o22; float wx2, wx1, wy2, wy1; };

__device__ __forceinline__ Sample make_sample(float x, float y) {
    int y1 = (int)floorf(y);
    int y2 = y1 + 1;
    int x1 = (int)floorf(x);
    int x2 = x1 + 1;
    y1 = min(max(y1, 0), FH - 1);
    y2 = min(max(y2, 0), FH - 1);
    x1 = min(max(x1, 0), FW - 1);
    x2 = min(max(x2, 0), FW - 1);
    Sample s;
    s.wy1 = y - (float)y1;
    s.wy2 = (float)y2 - y;
    s.wx1 = x - (float)x1;
    s.wx2 = (float)x2 - x;
    s.o11 = y1 * FW + x1;
    s.o12 = y1 * FW + x2;
    s.o21 = y2 * FW + x1;
    s.o22 = y2 * FW + x2;
    return s;
}

// Per-wave bin max: lanes stripe channels, gather 4 corners x 4 samples,
// running fmax, then wave32 reduction. Returns full reduced max in all lanes.
__device__ __forceinline__ float bin_max(const float* __restrict__ feat,
                                         const float* __restrict__ rois,
                                         int n, int iy, int ix, int lane) {
    const float r0 = rois[n * 4 + 0];   // y start
    const float r1 = rois[n * 4 + 1];   // x start
    const float r2 = rois[n * 4 + 2];   // y end
    const float r3 = rois[n * 4 + 3];   // x end
    const float sh = (r2 - r0) / 7.0f;  // exact: reference divides by R
    const float sw = (r3 - r1) / 7.0f;
    const float yb = r0 + sh * (float)iy;
    const float xb = r1 + sw * (float)ix;

    const float ysm0 = yb + sh / 3.0f;
    const float ysm1 = yb + 2.0f * sh / 3.0f;
    const float xsm0 = xb + sw / 3.0f;
    const float xsm1 = xb + 2.0f * sw / 3.0f;

    Sample sm[4];
    sm[0] = make_sample(xsm0, ysm0);
    sm[1] = make_sample(xsm1, ysm0);
    sm[2] = make_sample(xsm0, ysm1);
    sm[3] = make_sample(xsm1, ysm1);

    float m = -3.402823466e38f;
#pragma unroll 2
    for (int k = 0; k < CHN / 32; ++k) {
        const float* base = feat + (size_t)(k * 32 + lane) * (size_t)(FH * FW);
#pragma unroll
        for (int s = 0; s < 4; ++s) {
            const float f11 = base[sm[s].o11];
            const float f12 = base[sm[s].o12];
            const float f21 = base[sm[s].o21];
            const float f22 = base[sm[s].o22];
            // Same association as reference: p*wy2 + q*wy1.
            const float p = f11 * sm[s].wx2 + f12 * sm[s].wx1;
            const float q = f21 * sm[s].wx2 + f22 * sm[s].wx1;
            const float v = p * sm[s].wy2 + q * sm[s].wy1;
            m = fmaxf(m, v);
        }
    }
    return wave32_max(m);
}

// Phase 1: one wave32 per (roi, bin) -> pooled[n*49 + rr].
__global__ void roialign_pool_kernel(const float* __restrict__ feat,
                                     const float* __restrict__ rois,
                                     float* __restrict__ pooled,
                                     int n_bins) {
    const int wave = (int)((blockIdx.x * blockDim.x + threadIdx.x) >> 5);
    const int lane = (int)(threadIdx.x & 31u);
    if (wave >= n_bins) return;
    const int n  = wave / BINS_PER_ROI;
    const int rr = wave - n * BINS_PER_ROI;
    const int iy = rr / RS;
    const int ix = rr - iy * RS;

    const float m = bin_max(feat, rois, n, iy, ix, lane);
    if (lane == 0) pooled[wave] = m;
}

// Phase 2: broadcast pooled (N,49) -> out (N,C,7,7), coalesced NT stores
// (output has zero reuse; keep the 64MB feature map resident in L2 instead).
__global__ void roialign_bcast_kernel(const float* __restrict__ pooled,
                                      float* __restrict__ out,
                                      int total) {
    const int i = (int)(blockIdx.x * blockDim.x + threadIdx.x);
    if (i >= total) return;
    const unsigned u  = (unsigned)i;
    const unsigned n  = u / (unsigned)(CHN * BINS_PER_ROI);
    const unsigned rr = u % (unsigned)BINS_PER_ROI;
    __builtin_nontemporal_store(pooled[n * BINS_PER_ROI + rr], out + i);
}

// Fallback: fused single kernel (used only if workspace is too small).
__global__ void roialign_fused_kernel(const float* __restrict__ feat,
                                      const float* __restrict__ rois,
                                      float* __restrict__ out,
                                      int n_bins) {
    const int wave = (int)((blockIdx.x * blockDim.x + threadIdx.x) >> 5);
    const int lane = (int)(threadIdx.x & 31u);
    if (wave >= n_bins) return;
    const int n  = wave / BINS_PER_ROI;
    const int rr = wave - n * BINS_PER_ROI;
    const int iy = rr / RS;
    const int ix = rr - iy * RS;

    const float m = bin_max(feat, rois, n, iy, ix, lane);
#pragma unroll
    for (int k = 0; k < CHN / 32; ++k) {
        const int c = k * 32 + lane;
        __builtin_nontemporal_store(
            m, out + ((size_t)n * CHN + (size_t)c) * BINS_PER_ROI + rr);
    }
}

extern "C" void kernel_launch(void* const* d_in, const int* in_sizes, int n_in,
                              void* d_out, int out_size, void* d_ws, size_t ws_size,
                              hipStream_t stream) {
    const float* feat = (const float*)d_in[0];   // (C,H,W) f32
    const float* rois = (const float*)d_in[1];   // (N,4)   f32
    float* out = (float*)d_out;                  // (N,C,7,7) f32

    const int n_roi  = in_sizes[1] / 4;
    const int n_bins = n_roi * BINS_PER_ROI;

    const int threads = 256;                     // 8 wave32 per block
    const int waves_per_block = threads / 32;
    const int blocks1 = (n_bins + waves_per_block - 1) / waves_per_block;

    const size_t need = (size_t)n_bins * sizeof(float);
    if (ws_size >= need) {
        float* pooled = (float*)d_ws;
        roialign_pool_kernel<<<blocks1, threads, 0, stream>>>(feat, rois, pooled, n_bins);
        const int total = out_size;
        const int blocks2 = (total + threads - 1) / threads;
        roialign_bcast_kernel<<<blocks2, threads, 0, stream>>>(pooled, out, total);
    } else {
        roialign_fused_kernel<<<blocks1, threads, 0, stream>>>(feat, rois, out, n_bins);
    }
}